// Tripletlosshard1_54125177864860
// MI455X (gfx1250) — compile-verified
//
#include <hip/hip_runtime.h>

#define B_N 4096
#define L_N 2048
#define D_N 1024
#define LVL1 128
#define MARGIN 0.2f

typedef float v2f __attribute__((ext_vector_type(2)));
typedef float v8f __attribute__((ext_vector_type(8)));

// ---------------- accumulator init ----------------
__global__ void k_init(float* acc) {
    if (threadIdx.x < 4) acc[threadIdx.x] = 0.0f;
}

// ---------------- per-row inverse norms ----------------
// grid = B_N + L_N blocks of 256 threads; each thread covers 4 elements.
__global__ __launch_bounds__(256) void k_rnorm(const float* __restrict__ text,
                                               const float* __restrict__ label,
                                               float* __restrict__ rt,
                                               float* __restrict__ rl) {
    __shared__ float red[256];
    const int row = blockIdx.x;
    const float* src;
    float* dst;
    if (row < B_N) { src = text  + (size_t)row * D_N;          dst = &rt[row]; }
    else           { src = label + (size_t)(row - B_N) * D_N;  dst = &rl[row - B_N]; }
    float ss = 0.0f;
#pragma unroll
    for (int i = 0; i < 4; ++i) {
        float v = src[threadIdx.x + i * 256];
        ss += v * v;
    }
    red[threadIdx.x] = ss;
    __syncthreads();
    for (int s = 128; s > 0; s >>= 1) {
        if (threadIdx.x < s) red[threadIdx.x] += red[threadIdx.x + s];
        __syncthreads();
    }
    if (threadIdx.x == 0) {
        float nrm = sqrtf(red[0]);
        *dst = 1.0f / fmaxf(nrm, 1e-12f);
    }
}

// ---------------- similarity GEMM via V_WMMA_F32_16X16X4_F32 ----------------
// Register-blocked: each wave computes a 32x32 super-tile (2x2 WMMA tiles,
// 4 accumulators). Per K=4 step: 4 fragment loads -> 4 WMMAs (1 load/WMMA).
// 8 waves/block; (4096/32)*(2048/32)=8192 supertiles -> 1024 blocks.
__global__ __launch_bounds__(256) void k_sim_wmma(const float* __restrict__ text,
                                                  const float* __restrict__ label,
                                                  const float* __restrict__ rt,
                                                  const float* __restrict__ rl,
                                                  float* __restrict__ sim) {
    const int NT   = L_N / 32;                       // 64 n-supertiles
    const int wave = threadIdx.x >> 5;
    const int lane = threadIdx.x & 31;
    const int tile = blockIdx.x * 8 + wave;
    const int mt = tile / NT;
    const int nt = tile % NT;

    const int r  = lane & 15;
    const int kh = (lane >> 4) << 1;                 // 0 or 2 (K sub-offset per lane half)

    const float* __restrict__ a0 = text  + (size_t)(mt * 32 + r) * D_N + kh;
    const float* __restrict__ a1 = a0 + (size_t)16 * D_N;
    const float* __restrict__ b0 = label + (size_t)(nt * 32 + r) * D_N + kh;
    const float* __restrict__ b1 = b0 + (size_t)16 * D_N;

    v8f acc00 = {}, acc01 = {}, acc10 = {}, acc11 = {};
#pragma unroll 4
    for (int k = 0; k < D_N; k += 4) {
        v2f A0, A1, B0, B1;
        A0.x = a0[k]; A0.y = a0[k + 1];
        A1.x = a1[k]; A1.y = a1[k + 1];
        B0.x = b0[k]; B0.y = b0[k + 1];
        B1.x = b1[k]; B1.y = b1[k + 1];
        acc00 = __builtin_amdgcn_wmma_f32_16x16x4_f32(false, A0, false, B0, (short)0, acc00, false, false);
        acc01 = __builtin_amdgcn_wmma_f32_16x16x4_f32(false, A0, false, B1, (short)0, acc01, false, false);
        acc10 = __builtin_amdgcn_wmma_f32_16x16x4_f32(false, A1, false, B0, (short)0, acc10, false, false);
        acc11 = __builtin_amdgcn_wmma_f32_16x16x4_f32(false, A1, false, B1, (short)0, acc11, false, false);
    }

    // C/D layout: VGPR i -> M = i + 8*(lane>=16), N = lane&15
    const int ncol0 = nt * 32 + (lane & 15);
    const int ncol1 = ncol0 + 16;
    const int moff  = (lane >> 4) * 8;
    const float sl0 = rl[ncol0];
    const float sl1 = rl[ncol1];
#pragma unroll
    for (int i = 0; i < 8; ++i) {
        const int mrow0 = mt * 32 + moff + i;
        const int mrow1 = mrow0 + 16;
        const float st0 = rt[mrow0];
        const float st1 = rt[mrow1];
        sim[(size_t)mrow0 * L_N + ncol0] = acc00[i] * st0 * sl0;
        sim[(size_t)mrow0 * L_N + ncol1] = acc01[i] * st0 * sl1;
        sim[(size_t)mrow1 * L_N + ncol0] = acc10[i] * st1 * sl0;
        sim[(size_t)mrow1 * L_N + ncol1] = acc11[i] * st1 * sl1;
    }
}

// ---------------- per-row top-2 mining + triplet loss ----------------
// one block (128 threads) per row of sim.
__global__ __launch_bounds__(128) void k_loss(const float* __restrict__ sim,
                                              const int* __restrict__ target,
                                              float* __restrict__ accum) {
    __shared__ float sv1[2][128];
    __shared__ int   si1[2][128];
    __shared__ float sv2[2][128];

    const int row = blockIdx.x;
    const int tid = threadIdx.x;
    const float* __restrict__ srow = sim    + (size_t)row * L_N;
    const int*   __restrict__ trow = target + (size_t)row * L_N;

    const float NEG_INF = -__builtin_inff();
    float v1[2] = {NEG_INF, NEG_INF};
    float v2[2] = {NEG_INF, NEG_INF};
    int   i1[2] = {0x7fffffff, 0x7fffffff};

    for (int c = tid; c < L_N; c += 128) {
        const int lv = (c < LVL1) ? 0 : 1;
        const float s = srow[c];
        if (s > v1[lv])      { v2[lv] = v1[lv]; v1[lv] = s; i1[lv] = c; }
        else if (s > v2[lv]) { v2[lv] = s; }
    }
#pragma unroll
    for (int lv = 0; lv < 2; ++lv) { sv1[lv][tid] = v1[lv]; si1[lv][tid] = i1[lv]; sv2[lv][tid] = v2[lv]; }
    __syncthreads();

    for (int s = 64; s > 0; s >>= 1) {
        if (tid < s) {
#pragma unroll
            for (int lv = 0; lv < 2; ++lv) {
                const float av1 = sv1[lv][tid],     bv1 = sv1[lv][tid + s];
                const int   ai1 = si1[lv][tid],     bi1 = si1[lv][tid + s];
                const float av2 = sv2[lv][tid],     bv2 = sv2[lv][tid + s];
                // jax.lax.top_k tie-break: lower index first
                const bool aw = (av1 > bv1) || (av1 == bv1 && ai1 < bi1);
                sv1[lv][tid] = aw ? av1 : bv1;
                si1[lv][tid] = aw ? ai1 : bi1;
                sv2[lv][tid] = fmaxf(aw ? av2 : bv2, aw ? bv1 : av1);
            }
        }
        __syncthreads();
    }

    float tv1[2], tv2[2];
    int   ti1[2];
#pragma unroll
    for (int lv = 0; lv < 2; ++lv) { tv1[lv] = sv1[lv][0]; ti1[lv] = si1[lv][0]; tv2[lv] = sv2[lv][0]; }
    __syncthreads();

    // pass 2: loss terms. neg = (top1 index == p) ? top2 : top1
    // (covers both the has_hard and tie-fallback branches of the reference)
    float sum[2] = {0.0f, 0.0f};
    float cnt[2] = {0.0f, 0.0f};
    for (int c = tid; c < L_N; c += 128) {
        if (trow[c] > 0) {
            const int lv   = (c < LVL1) ? 0 : 1;
            const float sp = srow[c];
            const float ng = (ti1[lv] == c) ? tv2[lv] : tv1[lv];
            sum[lv] += fmaxf(MARGIN - sp + ng, 0.0f);
            cnt[lv] += 1.0f;
        }
    }
    sv1[0][tid] = sum[0]; sv1[1][tid] = sum[1];
    sv2[0][tid] = cnt[0]; sv2[1][tid] = cnt[1];
    __syncthreads();
    for (int s = 64; s > 0; s >>= 1) {
        if (tid < s) {
            sv1[0][tid] += sv1[0][tid + s]; sv1[1][tid] += sv1[1][tid + s];
            sv2[0][tid] += sv2[0][tid + s]; sv2[1][tid] += sv2[1][tid + s];
        }
        __syncthreads();
    }
    if (tid == 0) {
        atomicAdd(&accum[0], sv1[0][0]);   // sum level1
        atomicAdd(&accum[1], sv2[0][0]);   // cnt level1
        atomicAdd(&accum[2], sv1[1][0]);   // sum level2
        atomicAdd(&accum[3], sv2[1][0]);   // cnt level2
    }
}

// ---------------- finalize ----------------
__global__ void k_final(const float* __restrict__ accum, float* __restrict__ out) {
    const float l1 = accum[0] / fmaxf(accum[1], 1.0f);
    const float l2 = accum[2] / fmaxf(accum[3], 1.0f);
    out[0] = l1 + ((accum[3] >= 3.0f) ? l2 : 0.0f);
}

extern "C" void kernel_launch(void* const* d_in, const int* in_sizes, int n_in,
                              void* d_out, int out_size, void* d_ws, size_t ws_size,
                              hipStream_t stream) {
    const float* text   = (const float*)d_in[0];   // (4096, 1024)
    const float* label  = (const float*)d_in[1];   // (2048, 1024)
    const int*   target = (const int*)d_in[2];     // (4096, 2048)
    float* out = (float*)d_out;

    // workspace layout (floats): rt[4096] | rl[2048] | sim[4096*2048] | accum[4]
    float* ws    = (float*)d_ws;
    float* rt    = ws;
    float* rl    = rt + B_N;
    float* sim   = rl + L_N;
    float* accum = sim + (size_t)B_N * L_N;

    k_init<<<1, 32, 0, stream>>>(accum);
    k_rnorm<<<B_N + L_N, 256, 0, stream>>>(text, label, rt, rl);
    k_sim_wmma<<<(B_N / 32) * (L_N / 32) / 8, 256, 0, stream>>>(text, label, rt, rl, sim);
    k_loss<<<B_N, 128, 0, stream>>>(sim, target, accum);
    k_final<<<1, 1, 0, stream>>>(accum, out);
}